// MonteCarloDiscounted_7017976561893
// MI455X (gfx1250) — compile-verified
//
#include <hip/hip_runtime.h>
#include <stdint.h>

// Discounted Monte-Carlo return, reverse-time scan R[t] = r[t] + decay[b]*R[t+1].
// Memory-bound (AI ~= 0.25 flop/byte; roofline ~11.5us at 23.3 TB/s).
// 3-phase blocked linear scan:
//   P1: per-(chunk,column) local return with zero carry  (reward read #1, RT -> fills L2)
//   P2: tiny per-column reverse scan over chunk partials (carries stashed in d_out rows c*L)
//   P3: re-scan with carry-in, write outputs             (reward read #2, LU -> L2 hit+release;
//                                                         stores NT -> don't evict reward)
// Reward streaming uses CDNA5 async global->LDS copies (ASYNCcnt) with LDS double buffering.

#define GAMMA_F 0.99f

constexpr int LCHUNK  = 128;              // timesteps per chunk
constexpr int STAGE_T = 16;               // timesteps per LDS stage
constexpr int NSTAGE  = LCHUNK / STAGE_T; // 8
constexpr int BCOLS   = 256;              // columns per block (= blockDim.x)

// ---- CDNA5 async-copy helpers ------------------------------------------------

template <int N>
__device__ __forceinline__ void async_wait() {
#if __has_builtin(__builtin_amdgcn_s_wait_asynccnt)
  __builtin_amdgcn_s_wait_asynccnt(N);
#else
  asm volatile("s_wait_asynccnt %0" ::"n"(N) : "memory");
#endif
}

// per lane, copies 16 bytes from the global address gaddr to the LDS address lds_off; tracked by ASYNCcnt.
// LASTUSE=true marks the global read th:TH_LOAD_LU (last-use: NT + discard on hit)
// so phase 3 progressively releases L2 capacity it will never touch again.
template <bool LASTUSE>
__device__ __forceinline__ void async_load_b128(uint32_t lds_off, const void* gaddr) {
  if (LASTUSE)
    asm volatile("global_load_async_to_lds_b128 %0, %1, off th:TH_LOAD_LU"
                 ::"v"(lds_off), "v"(gaddr)
                 : "memory");
  else
    asm volatile("global_load_async_to_lds_b128 %0, %1, off"
                 ::"v"(lds_off), "v"(gaddr)
                 : "memory");
}

// Issue the async loads for one [STAGE_T x BCOLS] fp32 tile.
// 4 x b128 per thread -> exactly 4 ASYNCcnt increments per wave.
template <bool LASTUSE>
__device__ __forceinline__ void load_stage(const float* __restrict__ reward, int B,
                                           size_t tbase, int bx, int tid,
                                           uint32_t lds_buf_base) {
  const float* rowbase = reward + tbase * (size_t)B + (size_t)bx * BCOLS;
#pragma unroll
  for (int it = 0; it < 4; ++it) {
    int idx = it * BCOLS + tid;     // 0..1023 float4 slots in the tile
    int tt  = idx >> 6;             // tile row (64 float4 per 256-float row)
    int q   = idx & 63;             // float4 within row
    const void* g = (const void*)(rowbase + (size_t)tt * B + (size_t)q * 4);
    async_load_b128<LASTUSE>(lds_buf_base + (uint32_t)idx * 16u, g);
  }
}

// ---- Phase 1 / Phase 3 kernel (templated) ------------------------------------
// FINAL=false: compute chunk-local return A_c[b] (zero carry), store at out[c*L][b].
// FINAL=true : seed with carry-in from out[c*L][b], write all outputs (NT stores).

template <bool FINAL>
__global__ __launch_bounds__(BCOLS) void scan_chunk_kernel(
    const float* __restrict__ reward, const float* __restrict__ done,
    float* __restrict__ out, int B) {
  __shared__ float tile[2][STAGE_T * BCOLS];  // 2 x 16 KiB double buffer

  const int tid = threadIdx.x;
  const int bx  = blockIdx.x;
  const int c   = blockIdx.y;
  const int b   = bx * BCOLS + tid;
  const int t0  = c * LCHUNK;

  const float decay = (1.0f - done[b]) * GAMMA_F;

  // Generic LDS pointer -> 32-bit LDS byte offset (low 32 bits of the
  // shared-aperture generic address are the wave-relative LDS address).
  const uint32_t lbase0 = (uint32_t)(uintptr_t)(&tile[0][0]);
  const uint32_t lbase1 = (uint32_t)(uintptr_t)(&tile[1][0]);

  float acc;
  if (FINAL)
    acc = out[(size_t)t0 * B + b];  // carry-in written by chunk_carry_kernel
  else
    acc = 0.0f;

  // Prefetch the last (highest-t) stage.
  load_stage<FINAL>(reward, B, (size_t)(t0 + (NSTAGE - 1) * STAGE_T), bx, tid,
                    ((NSTAGE - 1) & 1) ? lbase1 : lbase0);

#pragma unroll
  for (int s = NSTAGE - 1; s >= 0; --s) {
    if (s > 0) {
      // Prefetch the next-older stage into the other buffer. The buffer
      // being overwritten was consumed in the previous iteration; the trailing
      // __syncthreads() of that iteration ordered all reads before this issue.
      load_stage<FINAL>(reward, B, (size_t)(t0 + (s - 1) * STAGE_T), bx, tid,
                        ((s - 1) & 1) ? lbase1 : lbase0);
      async_wait<4>();  // 4 outstanding = the just-issued stage; stage s landed
    } else {
      async_wait<0>();
    }
    __syncthreads();  // other waves' async loads also landed

    const float* buf = &tile[s & 1][0];
#pragma unroll
    for (int tt = STAGE_T - 1; tt >= 0; --tt) {
      float v = buf[tt * BCOLS + tid];
      acc = fmaf(decay, acc, v);  // R = r + decay*R
      if (FINAL) {
        // NT store: don't let the 128 MiB output evict reward from L2.
        __builtin_nontemporal_store(acc, &out[(size_t)(t0 + s * STAGE_T + tt) * B + b]);
      }
    }
    __syncthreads();  // protect buffer before it is refilled
  }

  if (!FINAL) out[(size_t)t0 * B + b] = acc;  // stash partial A_c[b] (re-read by P2: keep RT)
}

// ---- Phase 2: per-column reverse scan over chunk partials --------------------
// Reads A_c[b] from out[c*L][b]; replaces it with the carry-in for chunk c
// (i.e. R at the first timestep of chunk c+1).

__global__ void chunk_carry_kernel(float* __restrict__ out,
                                   const float* __restrict__ done, int B, int C) {
  int b = blockIdx.x * blockDim.x + threadIdx.x;
  if (b >= B) return;
  float decay = (1.0f - done[b]) * GAMMA_F;
  float dL = decay;
#pragma unroll
  for (int i = 0; i < 7; ++i) dL *= dL;  // decay^128 (LCHUNK = 2^7)

  float carry = 0.0f;  // R at first timestep of chunk c+1
  for (int c = C - 1; c >= 0; --c) {
    size_t off = (size_t)c * LCHUNK * B + b;
    float a    = out[off];   // A_c[b]
    out[off]   = carry;      // carry-in for chunk c (re-read by P3: keep RT)
    carry      = fmaf(dL, carry, a);  // Rstart_c = A_c + decay^L * Rstart_{c+1}
  }
}

// ---- Fallback for shapes not divisible by tile sizes -------------------------

__global__ void fallback_scan(const float* __restrict__ reward,
                              const float* __restrict__ done,
                              float* __restrict__ out, int T, int B) {
  int b = blockIdx.x * blockDim.x + threadIdx.x;
  if (b >= B) return;
  float decay = (1.0f - done[b]) * GAMMA_F;
  float acc = 0.0f;
  for (int t = T - 1; t >= 0; --t) {
    acc = fmaf(decay, acc, reward[(size_t)t * B + b]);
    out[(size_t)t * B + b] = acc;
  }
}

// ---- Launch ------------------------------------------------------------------

extern "C" void kernel_launch(void* const* d_in, const int* in_sizes, int n_in,
                              void* d_out, int out_size, void* d_ws, size_t ws_size,
                              hipStream_t stream) {
  const float* reward = (const float*)d_in[0];
  const float* done   = (const float*)d_in[1];
  float* out          = (float*)d_out;

  const int B = in_sizes[1];
  const int T = in_sizes[0] / B;

  if ((B % BCOLS == 0) && (T % LCHUNK == 0)) {
    const int C = T / LCHUNK;
    dim3 grid(B / BCOLS, C);
    scan_chunk_kernel<false><<<grid, BCOLS, 0, stream>>>(reward, done, out, B);
    chunk_carry_kernel<<<(B + BCOLS - 1) / BCOLS, BCOLS, 0, stream>>>(out, done, B, C);
    scan_chunk_kernel<true><<<grid, BCOLS, 0, stream>>>(reward, done, out, B);
  } else {
    fallback_scan<<<(B + BCOLS - 1) / BCOLS, BCOLS, 0, stream>>>(reward, done, out, T, B);
  }
}